// SparseEncoderSpatial_17970143167162
// MI455X (gfx1250) — compile-verified
//
#include <hip/hip_runtime.h>
#include <hip/hip_bf16.h>

typedef float v2f __attribute__((ext_vector_type(2)));
typedef float v8f __attribute__((ext_vector_type(8)));

#define Hh 256
#define Ww 256
#define Gg 8
#define NEDGE 32
#define FEATC 65
#define NPIX (Hh * Ww)
#define THRESH_F 15.0f
static const size_t FEATS_ELEMS = (size_t)Gg * NPIX * FEATC;  // 34,078,720

// ---------------------------------------------------------------------------
// Kernel 1: edge mask + pinfo via V_WMMA_F32_16X16X4_F32.
// A (16 edges x 4): per-edge affine coefficients over basis [v, u, 1, v^2+u^2]
// B (4 x 16 pixels): basis values for 16 consecutive pixels in one row
// D (16 edges x 16 pixels): lane = pixel column, VGPR r = edge row (+8 in hi half)
// ---------------------------------------------------------------------------
__global__ __launch_bounds__(256) void edge_wmma_kernel(
    const float* __restrict__ edges, float* __restrict__ out)
{
    const int g    = blockIdx.y;
    const int lane = threadIdx.x & 31;
    const int wave = threadIdx.x >> 5;
    const bool hi  = lane >= 16;
    const int n    = lane & 15;

    float* outFeats = out;
    float* outMask  = out + FEATS_ELEMS;

    // Per-lane A operands: 2 edge subtiles x 4 quantities.
    // A layout (f32 16x4): lanes 0-15 hold M=lane with K=0 (x), K=1 (y);
    // lanes 16-31 hold M=lane-16 with K=2 (x), K=3 (y).
    v2f And[2], Add[2], Ad0[2], Ad1[2];
#pragma unroll
    for (int s = 0; s < 2; ++s) {
        const int e = s * 16 + n;
        const float* ep = edges + ((g * NEDGE + e) << 2);
        const float e00 = ep[0] * 256.f, e01 = ep[1] * 256.f;
        const float e10 = ep[2] * 256.f, e11 = ep[3] * 256.f;
        const float d0 = e10 - e00, d1 = e11 - e01;
        const float L  = sqrtf(d0 * d0 + d1 * d1);
        const float Lc = fmaxf(L, 1e-4f);
        const float dn0 = d0 / Lc, dn1 = d1 / Lc;
        const float n0 = dn1, n1 = -dn0;              // normal
        const float cn = -(e00 * n0 + e01 * n1);
        const float rL = 1.f / Lc;
        const float dl0 = dn0 * rL, dl1 = dn1 * rL;
        const float cd = -(e00 * dn0 + e01 * dn1) * rL;
        const float c0 = e00 * e00 + e01 * e01;
        const float c1 = e10 * e10 + e11 * e11;

        v2f a;
        a.x = hi ? cn : n0;          a.y = hi ? 0.f : n1;          And[s] = a;
        a.x = hi ? cd : dl0;         a.y = hi ? 0.f : dl1;         Add[s] = a;
        a.x = hi ? c0 : -2.f * e00;  a.y = hi ? 1.f : -2.f * e01;  Ad0[s] = a;
        a.x = hi ? c1 : -2.f * e10;  a.y = hi ? 1.f : -2.f * e11;  Ad1[s] = a;
    }

    const int tileBase = blockIdx.x * 256 + wave * 32;
    for (int t = 0; t < 32; ++t) {
        const int T  = tileBase + t;          // 0..4095 global pixel tile
        const int v  = T >> 4;                // row
        const int ub = (T & 15) << 4;         // col base
        const float fv = (float)v;
        const float fu = (float)(ub + n);

        // B layout (f32 4x16): lanes 0-15 hold N=lane with K=0 (x), K=1 (y);
        // lanes 16-31 hold N=lane-16 with K=2 (x), K=3 (y).
        v2f B;
        B.x = hi ? 1.0f : fv;
        B.y = hi ? (fv * fv + fu * fu) : fu;

        v8f z = {};
        v8f Dnd[2], Ddd[2], Dq0[2], Dq1[2];
#pragma unroll
        for (int s = 0; s < 2; ++s) {
            Dnd[s] = __builtin_amdgcn_wmma_f32_16x16x4_f32(false, And[s], false, B, (short)0, z, false, false);
            Ddd[s] = __builtin_amdgcn_wmma_f32_16x16x4_f32(false, Add[s], false, B, (short)0, z, false, false);
            Dq0[s] = __builtin_amdgcn_wmma_f32_16x16x4_f32(false, Ad0[s], false, B, (short)0, z, false, false);
            Dq1[s] = __builtin_amdgcn_wmma_f32_16x16x4_f32(false, Ad1[s], false, B, (short)0, z, false, false);
        }

        float mf = 0.f, om = 0.f, anyf = 0.f;
#pragma unroll
        for (int s = 0; s < 2; ++s) {
#pragma unroll
            for (int r = 0; r < 8; ++r) {
                const float nd = fabsf(Dnd[s][r]);
                const float dd = Ddd[s][r];
                const bool m = ((nd <= THRESH_F) & (dd <= 1.f) & (dd >= 0.f))
                             | (Dq0[s][r] <= 225.f) | (Dq1[s][r] <= 225.f);
                const float off = fminf(dd, 1.f - dd);
                const float mfl = m ? 1.f : 0.f;
                mf += mfl;
                om += mfl * off;
                anyf = fmaxf(anyf, mfl);
            }
        }
        // Combine the two 8-edge row halves held by partner lanes (xor 16).
        mf  += __shfl_xor(mf, 16, 32);
        om  += __shfl_xor(om, 16, 32);
        anyf = fmaxf(anyf, __shfl_xor(anyf, 16, 32));

        if (lane < 16) {
            const int p = v * Ww + ub + lane;
            const size_t gp = (size_t)g * NPIX + p;
            outMask[gp] = anyf;                                  // edge_mask (0/1 float)
            outFeats[gp * FEATC + 64] = om / fmaxf(mf, 1e-4f);   // raw pinfo (masked in k2)
        }
    }
}

// ---------------------------------------------------------------------------
// Kernel 2: bilinear sample (degenerates to 0.25 * 4-neighbor sum with
// zero top/left border) + broadcast to 8 groups with masking.
// Block = one image row segment: v = blockIdx.y, u in [32*blockIdx.x, +32),
// all 64 channels. Streams 8 * 32 * 65 * 4 B of contiguous writes per block.
// ---------------------------------------------------------------------------
__global__ __launch_bounds__(256) void feats_kernel(
    const float* __restrict__ img, float* __restrict__ out)
{
    const int v  = blockIdx.y;
    const int ub = blockIdx.x * 32;
    const int t  = threadIdx.x;

    __shared__ float tile[32 * FEATC];   // [u_local][channel], slot 64 unused
    __shared__ float smMask[Gg][32];
    __shared__ float smPin[Gg][32];

    float* outFeats = out;
    const float* outMask = out + FEATS_ELEMS;

    {   // preload mask + raw pinfo for this pixel strip, all groups
        const int g = t >> 5, u_l = t & 31;
        const int p = v * Ww + ub + u_l;
        const size_t gp = (size_t)g * NPIX + p;
        smMask[g][u_l] = outMask[gp];
        smPin[g][u_l]  = outFeats[gp * FEATC + 64];
    }

    // sampled[c][u] = 0.25 * (I[v-1][u-1] + I[v-1][u] + I[v][u-1] + I[v][u])
#pragma unroll
    for (int base = 0; base < 2048; base += 256) {
        const int i   = base + t;
        const int u_l = i & 31;          // lane-fast -> coalesced row reads
        const int c   = i >> 5;
        const int u   = ub + u_l;
        const float* ch = img + (size_t)c * NPIX;
        const float a00 = (v > 0 && u > 0) ? ch[(v - 1) * Ww + (u - 1)] : 0.f;
        const float a01 = (v > 0)          ? ch[(v - 1) * Ww + u]       : 0.f;
        const float a10 = (u > 0)          ? ch[v * Ww + (u - 1)]       : 0.f;
        const float a11 =                    ch[v * Ww + u];
        tile[u_l * FEATC + c] = 0.25f * (a00 + a01 + a10 + a11);
    }
    __syncthreads();

#pragma unroll 1
    for (int g = 0; g < Gg; ++g) {
        const size_t gbase = ((size_t)g * NPIX + v * Ww + ub) * FEATC;
#pragma unroll
        for (int j = t; j < 32 * FEATC; j += 256) {   // j = u_l*65 + c, contiguous stores
            const int u_l = j / FEATC;
            const int c   = j - u_l * FEATC;
            const float val = (c < 64) ? tile[j] : smPin[g][u_l];
            outFeats[gbase + j] = val * smMask[g][u_l];
        }
    }
}

extern "C" void kernel_launch(void* const* d_in, const int* in_sizes, int n_in,
                              void* d_out, int out_size, void* d_ws, size_t ws_size,
                              hipStream_t stream) {
    const float* img   = (const float*)d_in[0];   // (1, 64, 256, 256) f32
    const float* edges = (const float*)d_in[1];   // (8, 32, 4) f32
    float* out = (float*)d_out;                   // feats (8,256,256,65) ++ edge_mask (8,256,256)

    // Kernel 1: 16 blocks x 8 groups; 8 waves/block, 32 pixel-tiles/wave.
    edge_wmma_kernel<<<dim3(16, Gg), 256, 0, stream>>>(edges, out);
    // Kernel 2: 8 u-strips x 256 rows.
    feats_kernel<<<dim3(Ww / 32, Hh), 256, 0, stream>>>(img, out);
}